// EarthMoversDistanceLoss_33895881900579
// MI455X (gfx1250) — compile-verified
//
#include <hip/hip_runtime.h>

// EMD loss: out[r] = sum_j ( cumsum(x[r]-y[r])[j] )^2
// Per-row prefix scan mapped onto chained V_WMMA_F32_16X16X4_F32:
//   Z (16x16) * U (upper-triangular ones), K=16 split into 4 chunks of 4.
// Each wave handles ROWS_PER_WAVE rows to amortize prologue / B-matrix setup.

typedef float v2f __attribute__((ext_vector_type(2)));
typedef float v8f __attribute__((ext_vector_type(8)));

#define BINS 256
#define WAVES_PER_BLOCK 8
#define ROWS_PER_WAVE 4

__device__ __forceinline__ float rdlane(float v, int srcLane) {
    // Uniform lane index -> v_readlane_b32 (SGPR result, no LDS traffic).
    return __int_as_float(__builtin_amdgcn_readlane(__float_as_int(v), srcLane));
}

__global__ __launch_bounds__(256) void emd_wmma_kernel(const float* __restrict__ x,
                                                       const float* __restrict__ y,
                                                       float* __restrict__ out,
                                                       int rows) {
    const int lane = threadIdx.x & 31;
    const int wave = threadIdx.x >> 5;
    int row = (blockIdx.x * WAVES_PER_BLOCK + wave) * ROWS_PER_WAVE;
    if (row >= rows) return;            // wave-uniform: EXEC all-ones for WMMA

    const int m    = lane & 15;         // A-matrix row / C-matrix column for this lane
    const int half = lane >> 4;         // lanes 16-31: K+2 (A/B), M+8 (C)

    // B chunks (4x16 each) of the upper-triangular ones matrix U[k][n] = (k <= n).
    // Row-invariant: built once per wave.
    v2f bmat[4];
    #pragma unroll
    for (int ck = 0; ck < 4; ++ck) {
        const int k0 = 4 * ck + 2 * half;
        bmat[ck][0] = (k0     <= m) ? 1.0f : 0.0f;
        bmat[ck][1] = (k0 + 1 <= m) ? 1.0f : 0.0f;
    }

    const int nrow = (rows - row) < ROWS_PER_WAVE ? (rows - row) : ROWS_PER_WAVE;
    for (int rr = 0; rr < nrow; ++rr, ++row) {
        const float* __restrict__ xr = x + (size_t)row * BINS;
        const float* __restrict__ yr = y + (size_t)row * BINS;

        v8f c = {0.f, 0.f, 0.f, 0.f, 0.f, 0.f, 0.f, 0.f};

        // D = Z * U accumulated over 4 K-chunks of the 16x16x16 product.
        #pragma unroll
        for (int ck = 0; ck < 4; ++ck) {
            // A chunk (16x4 f32): lane (m, half) holds Z[m][4*ck + 2*half + {0,1}]
            const int eoff = 16 * m + 4 * ck + 2 * half;
            v2f xa = *(const v2f*)(xr + eoff);   // global_load_b64, 8B aligned
            v2f ya = *(const v2f*)(yr + eoff);
            v2f a;
            a[0] = xa[0] - ya[0];                // -> v_pk_add_f32 (neg)
            a[1] = xa[1] - ya[1];

            c = __builtin_amdgcn_wmma_f32_16x16x4_f32(
                    /*neg_a=*/false, a, /*neg_b=*/false, bmat[ck],
                    /*c_mod=*/(short)0, c, /*reuse_a=*/false, /*reuse_b=*/false);
        }

        // C/D layout: VGPR r, lanes 0-15 -> P[r][lane]; lanes 16-31 -> P[8+r][lane-16].
        // Segment totals S[s] = P[s][15] live in lane 15 (s<8) and lane 31 (s>=8).
        float S[16];
        #pragma unroll
        for (int r = 0; r < 8; ++r) {
            S[r]     = rdlane(c[r], 15);
            S[r + 8] = rdlane(c[r], 31);
        }

        // Exclusive prefix of segment totals (wave-uniform scalar math).
        float O[16];
        O[0] = 0.0f;
        #pragma unroll
        for (int s = 1; s < 16; ++s) O[s] = O[s - 1] + S[s - 1];

        // Per-lane sum of squares of the full cumulative values.
        float acc = 0.0f;
        #pragma unroll
        for (int r = 0; r < 8; ++r) {
            const float o = half ? O[r + 8] : O[r];
            const float v = c[r] + o;
            acc = fmaf(v, v, acc);
        }

        // Wave32 tree reduction.
        #pragma unroll
        for (int off = 16; off > 0; off >>= 1)
            acc += __shfl_xor(acc, off, 32);

        if (lane == 0) out[row] = acc;
    }
}

extern "C" void kernel_launch(void* const* d_in, const int* in_sizes, int n_in,
                              void* d_out, int out_size, void* d_ws, size_t ws_size,
                              hipStream_t stream) {
    (void)n_in; (void)d_ws; (void)ws_size; (void)out_size;
    const float* x = (const float*)d_in[0];
    const float* y = (const float*)d_in[1];
    float* out = (float*)d_out;

    const int rows = in_sizes[0] / BINS;   // 262144
    const int rowsPerBlock = WAVES_PER_BLOCK * ROWS_PER_WAVE;

    dim3 block(256);
    dim3 grid((rows + rowsPerBlock - 1) / rowsPerBlock);
    hipLaunchKernelGGL(emd_wmma_kernel, grid, block, 0, stream, x, y, out, rows);
}